// GcnPromptLearner_65343632441953
// MI455X (gfx1250) — compile-verified
//
#include <hip/hip_runtime.h>

typedef float v2f __attribute__((ext_vector_type(2)));
typedef float v8f __attribute__((ext_vector_type(8)));

#define DIM 512
#define BM 128
#define BN 64
#define BK 32
#define LDA  (BK + 4)   // A row stride: 36 floats = 144B (16B aligned; b64 frag reads conflict-free)
#define LDB2 80         // B k-pair row stride in float2 units: 80*2 dwords == 32 (mod 64)
                        //  -> half-waves hit disjoint LDS bank halves on b64 fragment reads

// ---------------------------------------------------------------------------
// Dense GEMM  H[M x 512] = act(A[M x 512]) @ W[512 x 512]  via V_WMMA_F32_16X16X4_F32
// 256 threads = 8 waves (wave32). Block tile 128x64, K-panel 32 via LDS.
// B panel staged in fragment order (float2 {W[k][c], W[k+1][c]}) so each lane's
// B fragment is a single aligned ds_load_b64 into an even VGPR pair (no moves).
// ---------------------------------------------------------------------------
template <bool RELU>
__global__ __launch_bounds__(256) void gemm_wmma_f32(const float* __restrict__ A,
                                                     const float* __restrict__ W,
                                                     float* __restrict__ H,
                                                     int M) {
    __shared__ float  As[BM * LDA];
    __shared__ float2 Bs2[(BK / 2) * LDB2];

    const int tid   = threadIdx.x;
    const int lane  = tid & 31;
    const int wave  = tid >> 5;
    const int l15   = lane & 15;
    const int kHalf = lane >> 4;          // 0: lanes 0-15 (K=k,k+1), 1: lanes 16-31 (K=k+2,k+3)
    const int rowBase = blockIdx.y * BM;
    const int colBase = blockIdx.x * BN;

    v8f acc[4] = {};

    const int ar = wave * 16 + l15;

    for (int kBase = 0; kBase < DIM; kBase += BK) {
        __syncthreads();
        // ---- stage A tile: 128 x 32 floats (1024 float4, 4 per thread) ----
        for (int i = tid; i < BM * (BK / 4); i += 256) {
            int r  = i >> 3;           // / (BK/4)
            int c4 = i & 7;
            int gr = rowBase + r;
            float4 v = make_float4(0.f, 0.f, 0.f, 0.f);
            if (gr < M) {
                v = *(const float4*)(A + (size_t)gr * DIM + kBase + c4 * 4);
                if (RELU) {
                    v.x = fmaxf(v.x, 0.f); v.y = fmaxf(v.y, 0.f);
                    v.z = fmaxf(v.z, 0.f); v.w = fmaxf(v.w, 0.f);
                }
            }
            *(float4*)(&As[r * LDA + c4 * 4]) = v;
        }
        // ---- stage B tile, swizzled: Bs2[(k/2)*LDB2 + col] = {W[k][col], W[k+1][col]} ----
        for (int i = tid; i < BK * (BN / 4); i += 256) {
            int r  = i >> 4;           // k row 0..31
            int c4 = i & 15;           // float4 column group
            float4 v = *(const float4*)(W + (size_t)(kBase + r) * DIM + colBase + c4 * 4);
            float* dstp = (float*)&Bs2[(r >> 1) * LDB2] + (r & 1);
            dstp[(c4 * 4 + 0) * 2] = v.x;
            dstp[(c4 * 4 + 1) * 2] = v.y;
            dstp[(c4 * 4 + 2) * 2] = v.z;
            dstp[(c4 * 4 + 3) * 2] = v.w;
        }
        __syncthreads();

        // ---- compute: 8 K4-steps x 4 N-subtiles of V_WMMA_F32_16X16X4_F32 ----
#pragma unroll
        for (int kk2 = 0; kk2 < BK / 2; kk2 += 2) {       // kk2 = K/2; one WMMA covers 2 k-pairs
            v2f a = *(const v2f*)&As[ar * LDA + kk2 * 2 + kHalf * 2];
            const float2* brow = &Bs2[(kk2 + kHalf) * LDB2];
#pragma unroll
            for (int t = 0; t < 4; ++t) {
                v2f b = *(const v2f*)&brow[t * 16 + l15];
                acc[t] = __builtin_amdgcn_wmma_f32_16x16x4_f32(
                    /*neg_a=*/false, a, /*neg_b=*/false, b,
                    /*c_mod=*/(short)0, acc[t],
                    /*reuse_a=*/false, /*reuse_b=*/false);
            }
        }
    }

    // ---- store D: VGPR r -> row (wave*16 + r + 8*kHalf), col t*16 + l15 ----
#pragma unroll
    for (int t = 0; t < 4; ++t) {
#pragma unroll
        for (int r = 0; r < 8; ++r) {
            int gr = rowBase + wave * 16 + r + kHalf * 8;
            if (gr < M) H[(size_t)gr * DIM + colBase + t * 16 + l15] = acc[t][r];
        }
    }
}

// ---------------------------------------------------------------------------
// Degree / normalization helpers
// ---------------------------------------------------------------------------
__global__ void init_deg_kernel(float* deg1, float* deg2, int N) {
    int n = blockIdx.x * 256 + threadIdx.x;
    if (n < N) { deg1[n] = 1.0f; deg2[n] = 1.0f; }   // self-loop weight = 1
}

__global__ void deg_accum_kernel(const long long* __restrict__ dst,
                                 const float* __restrict__ ew,
                                 float* deg1, float* deg2, int E) {
    int e = blockIdx.x * 256 + threadIdx.x;
    if (e < E) {
        long long d = dst[e];
        unsafeAtomicAdd(&deg1[d], ew[e]);   // layer-1 weighted in-degree
        unsafeAtomicAdd(&deg2[d], 1.0f);    // layer-2 weight = 1
    }
}

__global__ void dinv_kernel(const float* __restrict__ deg, float* __restrict__ dinv, int N) {
    int n = blockIdx.x * 256 + threadIdx.x;
    if (n < N) {
        float d = deg[n];
        dinv[n] = (d > 0.f) ? rsqrtf(d) : 0.f;
    }
}

// out[n,f] = bias[f] + h[n,f] * dinv[n]^2   (self-loop message, norm = dinv*1*dinv)
__global__ void self_init_kernel(const float* __restrict__ h,
                                 const float* __restrict__ dinv,
                                 const float* __restrict__ bias,
                                 float* __restrict__ out, int total) {
    int i = blockIdx.x * 256 + threadIdx.x;
    if (i < total) {
        int n = i >> 9;           // / DIM
        int f = i & (DIM - 1);
        float di = dinv[n];
        out[i] = bias[f] + h[i] * di * di;
    }
}

// One wave per edge: out[dst] += h[src] * (dinv[src]*w*dinv[dst]), fp32 HW atomics (resolve in L2).
__global__ __launch_bounds__(256) void edge_scatter_kernel(const long long* __restrict__ src,
                                                           const long long* __restrict__ dst,
                                                           const float* __restrict__ ew,
                                                           const float* __restrict__ dinv,
                                                           const float* __restrict__ h,
                                                           float* __restrict__ out,
                                                           int E, int useW) {
    int e = blockIdx.x * 8 + (threadIdx.x >> 5);
    if (e >= E) return;
    int lane = threadIdx.x & 31;
    long long s = src[e];
    long long d = dst[e];
    const float4* hp = (const float4*)(h + (size_t)s * DIM);
    __builtin_prefetch(hp + lane, 0, 1);          // global_prefetch_b8: pull h[src] row while norm resolves
    float w = useW ? ew[e] : 1.0f;
    float norm = dinv[s] * w * dinv[d];
    float* op = out + (size_t)d * DIM;
#pragma unroll
    for (int i = 0; i < 4; ++i) {
        int idx = lane + i * 32;             // float4 index, 128 per row
        float4 v = hp[idx];
        unsafeAtomicAdd(op + idx * 4 + 0, v.x * norm);
        unsafeAtomicAdd(op + idx * 4 + 1, v.y * norm);
        unsafeAtomicAdd(op + idx * 4 + 2, v.z * norm);
        unsafeAtomicAdd(op + idx * 4 + 3, v.w * norm);
    }
}

// ---------------------------------------------------------------------------
// Launcher: 2-layer GCN
//   inputs: x[N,512], edge_index[2,E] (int64), edge_attr[E],
//           W1[512,512], b1[512], W2[512,512], b2[512]
// ---------------------------------------------------------------------------
extern "C" void kernel_launch(void* const* d_in, const int* in_sizes, int n_in,
                              void* d_out, int out_size, void* d_ws, size_t ws_size,
                              hipStream_t stream) {
    const float*     x   = (const float*)d_in[0];
    const long long* ei  = (const long long*)d_in[1];   // reference dtype: int64
    const float*     ew  = (const float*)d_in[2];
    const float*     W1  = (const float*)d_in[3];
    const float*     b1  = (const float*)d_in[4];
    const float*     W2  = (const float*)d_in[5];
    const float*     b2  = (const float*)d_in[6];
    float*           out = (float*)d_out;

    const int N = in_sizes[0] / DIM;       // 50000
    const int E = in_sizes[2];             // 160000
    const long long* src = ei;
    const long long* dst = ei + E;

    // workspace carve (all fp32, 16B-aligned offsets)
    float* h     = (float*)d_ws;            // N*DIM (reused by both layers)
    float* out1  = h    + (size_t)N * DIM;  // N*DIM
    float* deg1  = out1 + (size_t)N * DIM;  // N
    float* deg2  = deg1 + N;                // N
    float* dinv1 = deg2 + N;                // N
    float* dinv2 = dinv1 + N;               // N

    const int nBlkN  = (N + 255) / 256;
    const int nBlkE  = (E + 255) / 256;
    const int nBlkNF = (N * DIM) / 256;
    dim3 gemmGrid(DIM / BN, (N + BM - 1) / BM);   // (8, 391)

    // ---- degrees & symmetric norms for both layers ----
    init_deg_kernel<<<nBlkN, 256, 0, stream>>>(deg1, deg2, N);
    deg_accum_kernel<<<nBlkE, 256, 0, stream>>>(dst, ew, deg1, deg2, E);
    dinv_kernel<<<nBlkN, 256, 0, stream>>>(deg1, dinv1, N);
    dinv_kernel<<<nBlkN, 256, 0, stream>>>(deg2, dinv2, N);

    // ---- layer 1: h = x @ W1 ; out1 = b1 + self + scatter ----
    gemm_wmma_f32<false><<<gemmGrid, 256, 0, stream>>>(x, W1, h, N);
    self_init_kernel<<<nBlkNF, 256, 0, stream>>>(h, dinv1, b1, out1, N * DIM);
    edge_scatter_kernel<<<(E + 7) / 8, 256, 0, stream>>>(src, dst, ew, dinv1, h, out1, E, 1);

    // ---- layer 2: h = relu(out1) @ W2 ; out = b2 + self + scatter (weights = 1) ----
    gemm_wmma_f32<true><<<gemmGrid, 256, 0, stream>>>(out1, W2, h, N);
    self_init_kernel<<<nBlkNF, 256, 0, stream>>>(h, dinv2, b2, out, N * DIM);
    edge_scatter_kernel<<<(E + 7) / 8, 256, 0, stream>>>(src, dst, ew, dinv2, h, out, E, 0);
}